// BaseRouter_5841155523059
// MI455X (gfx1250) — compile-verified
//
#include <hip/hip_runtime.h>
#include <stdint.h>

// MoE router fused kernel for MI455X (gfx1250, wave32).
//   Pass 1: transpose W[4096,128] -> Wt[128,4096] in d_ws (2 MB, L2-resident).
//   Pass 2: logits = h[16384,4096] @ W  via V_WMMA_F32_16X16X4_F32,
//           fused gumbel top-8 / softmax / masked renorm epilogue.

typedef __attribute__((ext_vector_type(2))) float v2f;
typedef __attribute__((ext_vector_type(8))) float v8f;

#define T_TOK   16384
#define D_DIM   4096
#define E_EXP   128
#define TOPK    8
#define BR      64            // tokens per block
#define KC      64            // K-chunk
#define NCHUNK  (D_DIM / KC)  // 64
#define APITCH  68            // floats; 4m+k banks distinct, float4 aligned (272B)
#define BTPITCH 68            // Wt tile pitch: (k,k+1) contiguous -> ds_load_b64 frags
#define LPITCH  132           // epilogue logits tile pitch
#define NTHREADS 128          // 4 waves
#define BUFSZ   (BR * APITCH + E_EXP * BTPITCH)   // 13056 floats per buffer half

// ---------------- W transpose: W[d][e] -> Wt[e][d] ----------------
#define TS 32
__global__ __launch_bounds__(256)
void transpose_W(const float* __restrict__ W, float* __restrict__ Wt) {
    __shared__ float tile[TS][TS + 1];
    const int lx = threadIdx.x & 31;
    const int ly = threadIdx.x >> 5;          // 0..7
    const int d0 = blockIdx.x * TS;           // 128 blocks along D
    const int e0 = blockIdx.y * TS;           // 4 blocks along E
    #pragma unroll
    for (int r = ly; r < TS; r += 8)
        tile[r][lx] = W[(size_t)(d0 + r) * E_EXP + e0 + lx];
    __syncthreads();
    #pragma unroll
    for (int r = ly; r < TS; r += 8)
        Wt[(size_t)(e0 + r) * D_DIM + d0 + lx] = tile[lx][r];
}

__device__ __forceinline__ float hash01(uint32_t x) {
    x ^= x >> 17; x *= 0xed5ad4bbu;
    x ^= x >> 11; x *= 0xac4c1b51u;
    x ^= x >> 15; x *= 0x31848babu;
    x ^= x >> 14;
    return (float)(x >> 8) * (1.0f / 16777216.0f);
}

// ---------------- fused router ----------------
__global__ __launch_bounds__(NTHREADS)
void router_fused(const float* __restrict__ h, const float* __restrict__ Wt,
                  const unsigned char* __restrict__ tmask, float* __restrict__ out) {
    extern __shared__ float smem[];
    const int tid  = threadIdx.x;
    const int wave = tid >> 5;
    const int lane = tid & 31;
    const int row0 = blockIdx.x * BR;

    auto load_chunk = [&](int buf, int kc) {
        float* A  = smem + buf * BUFSZ;
        float* BT = smem + buf * BUFSZ + BR * APITCH;
        const float* hsrc = h + (size_t)row0 * D_DIM + kc * KC;
        #pragma unroll
        for (int i = 0; i < 8; ++i) {            // 64x64 A tile, float4 both sides
            int f  = tid + i * NTHREADS;
            int r  = f >> 4;
            int c4 = (f & 15) << 2;
            float4 v = *(const float4*)(hsrc + (size_t)r * D_DIM + c4);
            *(float4*)(A + r * APITCH + c4) = v;
        }
        const float* wsrc = Wt + kc * KC;        // Wt rows along k: coalesced
        #pragma unroll
        for (int i = 0; i < 16; ++i) {           // 128(e) x 64(k) tile, float4 both sides
            int f  = tid + i * NTHREADS;
            int e  = f >> 4;
            int c4 = (f & 15) << 2;
            float4 v = *(const float4*)(wsrc + (size_t)e * D_DIM + c4);
            *(float4*)(BT + e * BTPITCH + c4) = v;
        }
    };

    // ---- WMMA GEMM: each wave -> 16 rows x 128 experts (8 C tiles) ----
    v8f acc[8];
    #pragma unroll
    for (int j = 0; j < 8; ++j)
        #pragma unroll
        for (int q = 0; q < 8; ++q) acc[j][q] = 0.0f;

    load_chunk(0, 0);

    const int m    = lane & 15;            // A row within strip
    const int koff = (lane >> 4) << 1;     // lanes 16-31 hold K+2,K+3
    const int n    = lane & 15;            // B/C column within tile

    for (int c = 0; c < NCHUNK; ++c) {
        __syncthreads();
        if (c + 1 < NCHUNK) load_chunk((c + 1) & 1, c + 1);
        const int sel = c & 1;
        const float* A  = smem + sel * BUFSZ;
        const float* BT = smem + sel * BUFSZ + BR * APITCH;
        const float* aRow = A + (wave * 16 + m) * APITCH + koff;
        const float* bCol = BT + n * BTPITCH + koff;     // row (j*16+n), pair at +k0
        #pragma unroll
        for (int ks = 0; ks < KC / 4; ++ks) {
            const int k0 = ks * 4;
            v2f a = *(const v2f*)(aRow + k0);            // ds_load_b64: (k,k+1)
            #pragma unroll
            for (int j = 0; j < 8; ++j) {
                v2f b = *(const v2f*)(bCol + j * 16 * BTPITCH + k0);  // ds_load_b64
                acc[j] = __builtin_amdgcn_wmma_f32_16x16x4_f32(
                    false, a, false, b, (short)0, acc[j], false, false);
            }
        }
    }
    __syncthreads();

    // ---- stage logits tile to LDS (reuse buffer 0 region) ----
    float* L = smem;                           // 64 x LPITCH
    const int rbase = wave * 16 + ((lane >> 4) << 3);   // VGPR v: M=v (lo) / v+8 (hi)
    #pragma unroll
    for (int j = 0; j < 8; ++j)
        #pragma unroll
        for (int v = 0; v < 8; ++v)
            L[(rbase + v) * LPITCH + j * 16 + n] = acc[j][v];
    __syncthreads();

    const size_t TE = (size_t)T_TOK * E_EXP;
    float* out_mask  = out;
    float* out_probs = out + TE;
    float* out_clean = out + 2 * TE;
    float* out_sel   = out + 3 * TE;

    // coalesced logits_clean store
    #pragma unroll
    for (int i = 0; i < 16; ++i) {
        int f  = tid + i * NTHREADS;
        int r  = f >> 5;
        int c4 = (f & 31) << 2;
        float4 v = *(const float4*)(L + r * LPITCH + c4);
        *(float4*)(out_clean + (size_t)(row0 + r) * E_EXP + c4) = v;
    }

    // ---- per-token epilogue: one wave per token, 4 experts per lane ----
    const float NEG_INF  = -__builtin_inff();
    const float inv_tsel = 1.0f;   // 1/max(SELECT_TEMP or ROUTER_TEMP, 1e-6)
    const float inv_tmix = 1.0f;   // 1/max(ROUTER_TEMP, 1e-6)
    const float gtau     = 1.0f;   // GUMBEL_TAU

    for (int tt = wave; tt < BR; tt += 4) {
        const int t = row0 + tt;
        const bool active = tmask[t] != 0;

        float v[4], sel[4], p[4], sc[4];
        float mx = NEG_INF;
        #pragma unroll
        for (int i = 0; i < 4; ++i) {
            const int e = lane + 32 * i;
            v[i] = L[tt * LPITCH + e];
            float u = hash01((uint32_t)(t * E_EXP + e) * 2654435761u + 0x9e3779b9u);
            u = fminf(fmaxf(u, 1e-6f), 1.0f - 1e-6f);
            float g = -__logf(-__logf(u));
            float s = v[i] * inv_tsel + gtau * g;
            sel[i] = active ? s : NEG_INF;
            out_sel[(size_t)t * E_EXP + e] = sel[i];
            sc[i] = v[i] * inv_tmix;
            mx = fmaxf(mx, sc[i]);
        }
        #pragma unroll
        for (int o = 16; o; o >>= 1) mx = fmaxf(mx, __shfl_xor(mx, o, 32));
        float ssum = 0.0f;
        #pragma unroll
        for (int i = 0; i < 4; ++i) { p[i] = __expf(sc[i] - mx); ssum += p[i]; }
        #pragma unroll
        for (int o = 16; o; o >>= 1) ssum += __shfl_xor(ssum, o, 32);
        const float invs = 1.0f / ssum;

        // top-8 via 8 wave-wide argmax sweeps
        float ts[4];
        #pragma unroll
        for (int i = 0; i < 4; ++i) ts[i] = sel[i];
        unsigned selbits = 0;
        for (int kk = 0; kk < TOPK; ++kk) {
            float bv = NEG_INF; int bi = 0x7fffffff;
            #pragma unroll
            for (int i = 0; i < 4; ++i) {
                const int e = lane + 32 * i;
                if (ts[i] > bv) { bv = ts[i]; bi = e; }
            }
            #pragma unroll
            for (int o = 16; o; o >>= 1) {
                float ov = __shfl_xor(bv, o, 32);
                int   oi = __shfl_xor(bi, o, 32);
                if (ov > bv || (ov == bv && oi < bi)) { bv = ov; bi = oi; }
            }
            if (bi != 0x7fffffff && (bi & 31) == lane) {
                const int ii = bi >> 5;
                ts[ii] = NEG_INF;
                selbits |= (1u << ii);
            }
        }

        // masked renormalized probs
        float mp[4]; float d = 0.0f;
        #pragma unroll
        for (int i = 0; i < 4; ++i) {
            mp[i] = ((selbits >> i) & 1u) ? p[i] * invs : 0.0f;
            d += mp[i];
        }
        #pragma unroll
        for (int o = 16; o; o >>= 1) d += __shfl_xor(d, o, 32);
        const float invd = 1.0f / fmaxf(d, 1e-9f);
        #pragma unroll
        for (int i = 0; i < 4; ++i) {
            const int e = lane + 32 * i;
            const bool s = (((selbits >> i) & 1u) != 0u) && active;
            out_mask [(size_t)t * E_EXP + e] = s ? 1.0f : 0.0f;
            out_probs[(size_t)t * E_EXP + e] = active ? mp[i] * invd : 0.0f;
        }
    }
}

extern "C" void kernel_launch(void* const* d_in, const int* in_sizes, int n_in,
                              void* d_out, int out_size, void* d_ws, size_t ws_size,
                              hipStream_t stream) {
    const float* h = (const float*)d_in[0];
    const float* W = (const float*)d_in[1];
    const unsigned char* tm = (const unsigned char*)d_in[2];
    float* out = (float*)d_out;
    float* Wt  = (float*)d_ws;   // needs D*E*4 = 2 MB of workspace

    hipLaunchKernelGGL(transpose_W, dim3(D_DIM / TS, E_EXP / TS), dim3(256),
                       0, stream, W, Wt);

    dim3 grid(T_TOK / BR);      // 256 blocks
    dim3 block(NTHREADS);       // 4 waves
    size_t shmem = (size_t)2 * BUFSZ * sizeof(float); // ~102 KB
    hipLaunchKernelGGL(router_fused, grid, block, shmem, stream, h, Wt, tm, out);
}